// SequenceEncoder_noCNN_74792560492621
// MI455X (gfx1250) — compile-verified
//
#include <hip/hip_runtime.h>

// Problem constants (from the reference setup).
#define N_READS   2000
#define DIM       128
#define VOCAB     4
#define CHUNK     256        // tokens per thread in the histogram pass (multiple of 4)

typedef __attribute__((ext_vector_type(2))) float v2f;
typedef __attribute__((ext_vector_type(8))) float v8f;
typedef __attribute__((ext_vector_type(4))) int   v4i;

// ---------------------------------------------------------------------------
// Pass 0: zero the counts workspace (harness poisons d_ws with 0xAA).
// ---------------------------------------------------------------------------
__global__ void seqenc_zero_counts(unsigned int* __restrict__ counts, int n) {
    int i = blockIdx.x * blockDim.x + threadIdx.x;
    if (i < n) counts[i] = 0u;
}

// ---------------------------------------------------------------------------
// Flush packed 4x16-bit per-thread counters into the global histogram.
// ---------------------------------------------------------------------------
__device__ __forceinline__ void flush_counts(unsigned int* __restrict__ counts,
                                             int seg, unsigned long long& pack) {
#pragma unroll
    for (int v = 0; v < VOCAB; ++v) {
        unsigned int cv = (unsigned int)((pack >> (v * 16)) & 0xFFFFull);
        if (cv) atomicAdd(&counts[seg * VOCAB + v], cv);
    }
    pack = 0ull;
}

// ---------------------------------------------------------------------------
// Pass 1: per-segment vocab histogram (bandwidth-dominant: 16 MB streamed).
//  - b128 non-temporal loads: 4 tokens + 4 segids per VMEM request.
//  - packed 64-bit counter (tok -> +1<<(tok*16)): branch-free, no dynamic
//    register indexing, no scratch.
//  - segment_ids is sorted, so a 256-token chunk crosses ~1-2 boundaries ->
//    only a handful of global atomics per thread.
// ---------------------------------------------------------------------------
__global__ void seqenc_histogram(const int* __restrict__ tokens,
                                 const int* __restrict__ segids,
                                 unsigned int* __restrict__ counts,
                                 int total) {
    int tid = blockIdx.x * blockDim.x + threadIdx.x;
    long start = (long)tid * CHUNK;
    if (start >= total) return;
    long end = start + CHUNK;
    if (end > total) end = total;

    unsigned long long pack = 0ull;   // 4 x 16-bit counters
    int cur = segids[start];

    long t = start;
    for (; t + 4 <= end; t += 4) {
        v4i s4 = __builtin_nontemporal_load((const v4i*)(segids + t));
        v4i k4 = __builtin_nontemporal_load((const v4i*)(tokens + t));
#pragma unroll
        for (int j = 0; j < 4; ++j) {
            int sj = s4[j];
            if (sj != cur) {                 // rare: segment boundary
                flush_counts(counts, cur, pack);
                cur = sj;
            }
            pack += 1ull << ((k4[j] & (VOCAB - 1)) * 16);
        }
    }
    for (; t < end; ++t) {                   // scalar tail
        int sj = segids[t];
        if (sj != cur) {
            flush_counts(counts, cur, pack);
            cur = sj;
        }
        pack += 1ull << ((tokens[t] & (VOCAB - 1)) * 16);
    }
    flush_counts(counts, cur, pack);
}

// ---------------------------------------------------------------------------
// Pass 2: out[16x16 tile] = (counts[16x4] * inv_total) x E[4x16] via
// V_WMMA_F32_16X16X4_F32.  One wave per (row-tile, col-tile):
//   grid.x = 2000/16 = 125 row tiles, 8 waves/block = 8 col tiles (128/16).
//
// A-matrix 16x4 f32 layout (ISA 7.12.2): lanes 0-15 -> M=lane, v0=K0, v1=K1;
//                                        lanes 16-31 -> M=lane-16, v0=K2, v1=K3.
// B-matrix 4x16 f32 (transpose-symmetric): lane&15 -> N, v0=K(2*half), v1=K(2*half+1).
// C/D 16x16 f32: lane&15 -> N; VGPR j -> M = j + 8*(lane>=16).
// ---------------------------------------------------------------------------
__global__ void seqenc_combine_wmma(const unsigned int* __restrict__ counts,
                                    const float* __restrict__ emb,   // [VOCAB][DIM]
                                    float* __restrict__ out) {       // [N_READS][DIM]
    const int lane  = threadIdx.x & 31;
    const int wave  = threadIdx.x >> 5;        // 0..7 -> column tile
    const int rowB  = blockIdx.x * 16;
    const int colB  = wave * 16;

    const int m     = lane & 15;
    const int khalf = lane >> 4;               // 0: K={0,1}, 1: K={2,3}
    const int row   = rowB + m;

    // Row histogram + normalization (empty segment -> inv = 0 -> zero output,
    // matching reference's sums/max(count,1) behavior).
    const unsigned int c0 = counts[row * 4 + 0];
    const unsigned int c1 = counts[row * 4 + 1];
    const unsigned int c2 = counts[row * 4 + 2];
    const unsigned int c3 = counts[row * 4 + 3];
    const float tot = (float)(c0 + c1 + c2 + c3);
    const float inv = (tot > 0.0f) ? (1.0f / tot) : 0.0f;

    v2f a;
    a.x = (khalf ? (float)c2 : (float)c0) * inv;
    a.y = (khalf ? (float)c3 : (float)c1) * inv;

    const int n = lane & 15;
    v2f b;
    b.x = emb[(khalf * 2 + 0) * DIM + colB + n];
    b.y = emb[(khalf * 2 + 1) * DIM + colB + n];

    v8f acc = {};
    acc = __builtin_amdgcn_wmma_f32_16x16x4_f32(
        /*neg_a=*/false, a, /*neg_b=*/false, b,
        /*c_mod=*/(short)0, acc, /*reuse_a=*/false, /*reuse_b=*/false);

    const int mB = khalf * 8;
#pragma unroll
    for (int j = 0; j < 8; ++j) {
        __builtin_nontemporal_store(acc[j], &out[(rowB + mB + j) * DIM + colB + n]);
    }
}

// ---------------------------------------------------------------------------
extern "C" void kernel_launch(void* const* d_in, const int* in_sizes, int n_in,
                              void* d_out, int out_size, void* d_ws, size_t ws_size,
                              hipStream_t stream) {
    const int*   tokens = (const int*)d_in[0];
    const int*   segids = (const int*)d_in[1];
    const float* emb    = (const float*)d_in[2];
    const int    total  = in_sizes[0];

    unsigned int* counts = (unsigned int*)d_ws;   // [N_READS][VOCAB] u32 = 32 KB
    const int ncnt = N_READS * VOCAB;

    seqenc_zero_counts<<<(ncnt + 255) / 256, 256, 0, stream>>>(counts, ncnt);

    const int nthreads = (total + CHUNK - 1) / CHUNK;
    seqenc_histogram<<<(nthreads + 255) / 256, 256, 0, stream>>>(
        tokens, segids, counts, total);

    seqenc_combine_wmma<<<N_READS / 16, 256, 0, stream>>>(
        counts, emb, (float*)d_out);
}